// DarkCLoss_10462540333743
// MI455X (gfx1250) — compile-verified
//
#include <hip/hip_runtime.h>
#include <hip/hip_bf16.h>

// B=16, C=3, H=W=512, 35x35 separable sliding max of (1-x) channel-max,
// loss = -mean(dc).  Memory-bound: 50 MB input; intermediates L2-resident.
//
// Pipeline (deterministic, no atomics):
//   k1 chanmax     : y = 1 - min_c(x)                  (float4 loads)
//   k2 vmax35_async: vertical 35-max; 42x256 tile staged in LDS via
//                    global_load_async_to_lds_b128 (ASYNCcnt), row-clamped
//   k3 hmax35sum   : horizontal 35-max + per-block partial sums
//                    (wave reduction via V_WMMA_F32_16X16X4_F32)
//   k4 finalize    : fixed-order reduce of 2048 partials, out = -sum/N

#define BATCH 16
#define HDIM  512
#define WDIM  512
#define HW    (HDIM * WDIM)          // 262144
#define NPIX  (BATCH * HW)           // 4194304
#define NEGBIG (-3.402823466e+38f)

typedef __attribute__((ext_vector_type(2))) float v2f;
typedef __attribute__((ext_vector_type(8))) float v8f;

// ---- wave32 sum reduction via V_WMMA_F32_16X16X4_F32 ----------------------
// A(16x4): lane L<16 holds A[L][0..1]=s(L); lane L>=16 holds A[L-16][2..3]=s(L).
// B(4x16): per-lane (1,0) -> B[0]=B[2]=1, B[1]=B[3]=0.
// D[m][n] = s(m) + s(m+16); lane n sums its 8 accs; shfl_xor(16) folds halves.
__device__ __forceinline__ float wave_sum32(float s) {
#if __has_builtin(__builtin_amdgcn_wmma_f32_16x16x4_f32)
  v2f a; a[0] = s; a[1] = s;
  v2f b; b[0] = 1.0f; b[1] = 0.0f;
  v8f c = {};
  c = __builtin_amdgcn_wmma_f32_16x16x4_f32(false, a, false, b,
                                            (short)0, c, false, false);
  float r = ((c[0] + c[1]) + (c[2] + c[3])) + ((c[4] + c[5]) + (c[6] + c[7]));
  r += __shfl_xor(r, 16, 32);
  return r;
#else
  for (int m = 16; m > 0; m >>= 1) s += __shfl_xor(s, m, 32);
  return s;
#endif
}

// ---- async-copy helpers (CDNA5 Tensor/async path, ASYNCcnt) ---------------
__device__ __forceinline__ void async_g2lds_b128(unsigned lds_byte_addr,
                                                 unsigned gbyte_off,
                                                 unsigned long long sbase) {
  // GVS mode: mem_addr = SGPR base + 32-bit VGPR offset; LDS addr from VDST.
  asm volatile("global_load_async_to_lds_b128 %0, %1, %2"
               :: "v"(lds_byte_addr), "v"(gbyte_off), "s"(sbase)
               : "memory");
}

__device__ __forceinline__ void wait_async0() {
#if __has_builtin(__builtin_amdgcn_s_wait_asynccnt)
  __builtin_amdgcn_s_wait_asynccnt(0);
#else
  asm volatile("s_wait_asynccnt 0x0" ::: "memory");
#endif
}

// ---- k1: channel max of (1 - x) ------------------------------------------
__global__ void k_chanmax(const float* __restrict__ x, float* __restrict__ y) {
  int i = blockIdx.x * blockDim.x + threadIdx.x;   // over NPIX/4
  int b = i >> 16;                                 // HW/4 = 65536
  int r = i & 65535;
  const float4* x4 = (const float4*)x;
  float4* y4 = (float4*)y;
  size_t base = (size_t)b * (3 * HW / 4) + r;
  float4 c0 = x4[base];
  float4 c1 = x4[base + HW / 4];
  float4 c2 = x4[base + 2 * (HW / 4)];
  float4 o;
  o.x = 1.0f - fminf(c0.x, fminf(c1.x, c2.x));
  o.y = 1.0f - fminf(c0.y, fminf(c1.y, c2.y));
  o.z = 1.0f - fminf(c0.z, fminf(c1.z, c2.z));
  o.w = 1.0f - fminf(c0.w, fminf(c1.w, c2.w));
  y4[(size_t)b * (HW / 4) + r] = o;
}

// ---- 35-window max over v[0..41] producing 8 outputs ----------------------
// output j (j=0..7) covers k in [j, j+34]; common core k in [7,34].
__device__ __forceinline__ void max35x8(const float v[42], float out[8]) {
  float common = v[7];
#pragma unroll
  for (int k = 8; k <= 34; ++k) common = fmaxf(common, v[k]);
  float S[7];
  S[6] = v[6];
#pragma unroll
  for (int j = 5; j >= 0; --j) S[j] = fmaxf(v[j], S[j + 1]);
  float P[7];
  P[0] = v[35];
#pragma unroll
  for (int j = 1; j < 7; ++j) P[j] = fmaxf(P[j - 1], v[35 + j]);
  out[0] = fmaxf(common, S[0]);
#pragma unroll
  for (int j = 1; j < 7; ++j) out[j] = fmaxf(fmaxf(common, S[j]), P[j - 1]);
  out[7] = fmaxf(common, P[6]);
}

// ---- k2: vertical 35-max, LDS tile staged by async DMA --------------------
// Block = (b, 8-row chunk, 256-column half).  Tile = 42 rows x 256 cols.
// Row indices are clamped to [0,511]: a clamped tap duplicates an edge row
// that is provably inside the true window whenever clamping occurs, so the
// result equals the -inf-padded reference.
#define TCOLS 256
__global__ void k_vmax35_async(const float* __restrict__ y,
                               float* __restrict__ t) {
  __shared__ float tile[42 * TCOLS];              // 43008 B
  int tid = threadIdx.x;
  int blk = blockIdx.x;                           // 16*64*2 = 2048
  int whalf = blk & 1;
  int chunk = (blk >> 1) & 63;
  int b = blk >> 7;
  int h0 = chunk << 3;
  int w0 = whalf << 8;

  unsigned long long ybase = (unsigned long long)y;
  // 42 rows * 64 float4 = 2688 16-byte slots, spread over 256 threads
  for (int s = tid; s < 42 * 64; s += 256) {
    int r = s >> 6;
    int c4 = (s & 63) << 2;                       // float index within row
    int gr = h0 - 17 + r;
    gr = min(max(gr, 0), HDIM - 1);               // clamp == -inf padding
    unsigned goff = (unsigned)(((size_t)b * HW + (size_t)gr * WDIM + w0 + c4) * 4u);
    unsigned loff = (unsigned)(size_t)&tile[(r << 8) + c4];
    async_g2lds_b128(loff, goff, ybase);
  }
  wait_async0();
  __syncthreads();

  float v[42];
#pragma unroll
  for (int k = 0; k < 42; ++k) v[k] = tile[(k << 8) + tid];
  float o[8];
  max35x8(v, o);

  float* tb = t + (size_t)b * HW + w0 + tid;
#pragma unroll
  for (int j = 0; j < 8; ++j) tb[(size_t)(h0 + j) * WDIM] = o[j];
}

// ---- k3: horizontal 35-max + block partial sums ---------------------------
__global__ void k_hmax35_sum(const float* __restrict__ t,
                             float* __restrict__ partials) {
  int tid = blockIdx.x * blockDim.x + threadIdx.x;   // 16*512*64 threads
  int wc = tid & 63;
  int h = (tid >> 6) & 511;
  int b = tid >> 15;
  int w0 = wc << 3;
  const float* row = t + (size_t)b * HW + (size_t)h * WDIM;
  float v[42];
#pragma unroll
  for (int k = 0; k < 42; ++k) {
    int c = w0 - 17 + k;
    v[k] = (c >= 0 && c < WDIM) ? row[c] : NEGBIG;
  }
  float o[8];
  max35x8(v, o);
  float lsum = ((o[0] + o[1]) + (o[2] + o[3])) + ((o[4] + o[5]) + (o[6] + o[7]));

  // wave32 reduction via WMMA (EXEC all-ones here: exact grid, no divergence)
  float wsum = wave_sum32(lsum);

  __shared__ float acc[8];
  int lane = threadIdx.x & 31;
  int wave = threadIdx.x >> 5;
  if (lane == 0) acc[wave] = wsum;
  __syncthreads();
  if (threadIdx.x == 0) {
    float s = acc[0];
#pragma unroll
    for (int i = 1; i < 8; ++i) s += acc[i];
    partials[blockIdx.x] = s;   // fixed slot -> deterministic
  }
}

// ---- k4: deterministic final reduce of 2048 partials ----------------------
__global__ void k_finalize(const float* __restrict__ partials,
                           float* __restrict__ out) {
  __shared__ float buf[256];
  float s = 0.0f;
#pragma unroll
  for (int i = 0; i < 8; ++i) s += partials[threadIdx.x + i * 256];
  buf[threadIdx.x] = s;
  __syncthreads();
  for (int off = 128; off > 0; off >>= 1) {
    if ((int)threadIdx.x < off) buf[threadIdx.x] += buf[threadIdx.x + off];
    __syncthreads();
  }
  if (threadIdx.x == 0) out[0] = -buf[0] / (float)NPIX;
}

extern "C" void kernel_launch(void* const* d_in, const int* in_sizes, int n_in,
                              void* d_out, int out_size, void* d_ws, size_t ws_size,
                              hipStream_t stream) {
  (void)in_sizes; (void)n_in; (void)out_size; (void)ws_size;
  const float* x = (const float*)d_in[0];
  float* out = (float*)d_out;

  float* ws = (float*)d_ws;
  float* partials = ws;                 // 2048 floats
  float* y = ws + 4096;                 // NPIX floats (16-KB aligned offset)
  float* t = y + NPIX;                  // NPIX floats  (~33.6 MB total)

  k_chanmax      <<<NPIX / 4 / 256, 256, 0, stream>>>(x, y);
  k_vmax35_async <<<BATCH * 64 * 2, 256, 0, stream>>>(y, t);
  k_hmax35_sum   <<<(BATCH * HDIM * 64) / 256, 256, 0, stream>>>(t, partials);
  k_finalize     <<<1, 256, 0, stream>>>(partials, out);
}